// DecorrLoss_2224793059686
// MI455X (gfx1250) — compile-verified
//
#include <hip/hip_runtime.h>

// DecorrLoss forward on MI455X (gfx1250, wave32).
// Two 128x128 Gram matrices via V_WMMA_F32_16X16X4_F32:
//   Gs = X^T X, H = (X^2)^T (X^2)
// Input staged HBM->LDS by the Tensor Data Mover (tensor_load_to_lds) with
// hardware row padding (128 DW rows + 4 DW pad = 132-float stride), double
// buffered so the DMA of tile kt+1 overlaps WMMA of tile kt.

typedef __attribute__((ext_vector_type(2))) float v2f;
typedef __attribute__((ext_vector_type(8))) float v8f;
typedef __attribute__((ext_vector_type(4))) unsigned int u32x4;
typedef __attribute__((ext_vector_type(4))) int i32x4;
typedef __attribute__((ext_vector_type(8))) int i32x8;

#define D          128
#define NROWS      32768          // 4 * 8192
#define CHUNK      512            // rows per block
#define KTILE      32             // rows per LDS tile
#define LDS_STRIDE 132            // 128 + 4 pad floats (TDM pad) per row
#define TILE_ELEMS (KTILE * LDS_STRIDE)
#define KAPPA      0.5f

__global__ void decorr_zero_kernel(float* __restrict__ p, int n) {
  int i = blockIdx.x * blockDim.x + threadIdx.x;
  if (i < n) p[i] = 0.0f;
}

// Issue one TDM load: 32x128 fp32 tile at `gaddr` -> LDS offset `lds_addr`,
// padded to 132-float rows. Wave-level op; EXEC ignored; tracked by TENSORcnt.
__device__ __forceinline__ void tdm_load_tile(unsigned lds_addr,
                                              unsigned long long gaddr) {
  u32x4 g0;
  g0.x = 1u;                                            // count=1, user mode
  g0.y = lds_addr;                                      // LDS byte address
  g0.z = (unsigned)gaddr;                               // global_addr[31:0]
  g0.w = (unsigned)((gaddr >> 32) & 0x1FFFFFFull)       // global_addr[56:32]
         | (2u << 30);                                  // type=2 ("image")
  // data_size=4B(2<<16) | pad_enable(1<<20) | pad_interval=6:128DW(6<<22)
  // | pad_amount=3:4DW(3<<25)
  i32x8 g1 = {(int)((2u << 16) | (1u << 20) | (6u << 22) | (3u << 25)),
              (int)(128u << 16),   // tensor_dim0 = 128 (low 16 in [63:48])
              (int)(32u << 16),    // tensor_dim1 = 32  (low 16 in [95:80])
              (int)(128u << 16),   // tile_dim0   = 128 ([127:112])
              32,                  // tile_dim1   = 32  ([143:128])
              128,                 // tensor_dim0_stride = 128 ([191:160])
              0, 0};
  i32x4 z4 = {0, 0, 0, 0};
#if defined(__clang_major__) && __clang_major__ >= 23
  i32x8 z8 = {0, 0, 0, 0, 0, 0, 0, 0};
  __builtin_amdgcn_tensor_load_to_lds(g0, g1, z4, z4, z8, 0);
#else
  __builtin_amdgcn_tensor_load_to_lds(g0, g1, z4, z4, 0);
#endif
}

__global__ __launch_bounds__(256) void decorr_gram_kernel(
    const float* __restrict__ x, float* __restrict__ gs, float* __restrict__ h) {
  __shared__ float lds[2 * TILE_ELEMS];                 // double buffer, 33 KB

  const int tid  = threadIdx.x;
  const int lane = tid & 31;       // wave32
  const int wave = tid >> 5;       // 0..7 -> i-tile owned by this wave
  const int half = lane >> 4;      // K half-select for A/B fragments
  const int m    = lane & 15;      // M (A) / N (B) index within tile

  v8f accG[8], accH[8];            // 8 j-tiles x {Gs,H}: 128 accumulator VGPRs
#pragma unroll
  for (int t = 0; t < 8; ++t) { accG[t] = {}; accH[t] = {}; }

  const long n0 = (long)blockIdx.x * CHUNK;
  const int TILES = CHUNK / KTILE;                      // 16
  // Low 32 bits of a flat LDS pointer are the LDS byte offset (aperture is in
  // the upper 32 bits on gfx1250).
  const unsigned lds_base = (unsigned)(unsigned long long)(uintptr_t)&lds[0];
  const unsigned long long xbase =
      (unsigned long long)(uintptr_t)(x + n0 * D);

  if (wave == 0)                                        // prime tile 0 DMA
    tdm_load_tile(lds_base, xbase);

  for (int kt = 0; kt < TILES; ++kt) {
    const int buf = kt & 1;
    __syncthreads();   // all waves done reading buf^1 (compute of kt-1)
    if (wave == 0) {
      if (kt + 1 < TILES) {
        tdm_load_tile(lds_base + (unsigned)((buf ^ 1) * TILE_ELEMS * 4),
                      xbase + (unsigned long long)(kt + 1) * KTILE * D * 4ull);
        __builtin_amdgcn_s_wait_tensorcnt(1);           // tile kt landed
      } else {
        __builtin_amdgcn_s_wait_tensorcnt(0);
      }
    }
    __syncthreads();   // tile kt visible to all waves
    const float* tile = &lds[buf * TILE_ELEMS];

    for (int k = 0; k < KTILE; k += 4) {
      // A fragment (16x4 f32): lanes 0-15 K={k,k+1}, lanes 16-31 K={k+2,k+3}
      const int r0 = (k + 2 * half) * LDS_STRIDE;
      const int r1 = r0 + LDS_STRIDE;
      v2f a;
      a.x = tile[r0 + wave * 16 + m];
      a.y = tile[r1 + wave * 16 + m];
      v2f ay = a * a;                // X^2 fragment = elementwise square

#pragma unroll
      for (int t = 0; t < 8; ++t) {
        v2f b;                       // B fragment (4x16), mirror layout of A
        b.x = tile[r0 + t * 16 + m];
        b.y = tile[r1 + t * 16 + m];
        v2f by = b * b;
        accG[t] = __builtin_amdgcn_wmma_f32_16x16x4_f32(
            false, a,  false, b,  (short)0, accG[t], false, false);
        accH[t] = __builtin_amdgcn_wmma_f32_16x16x4_f32(
            false, ay, false, by, (short)0, accH[t], false, false);
      }
    }
  }

  // C/D layout: VGPR r -> M = r + 8*half, N = lane&15
#pragma unroll
  for (int t = 0; t < 8; ++t) {
#pragma unroll
    for (int r = 0; r < 8; ++r) {
      const int i = wave * 16 + r + 8 * half;
      const int j = t * 16 + m;
      atomicAdd(&gs[i * D + j], accG[t][r]);
      atomicAdd(&h [i * D + j], accH[t][r]);
    }
  }
}

__global__ __launch_bounds__(256) void decorr_finalize_kernel(
    const float* __restrict__ gs, const float* __restrict__ h,
    float* __restrict__ out) {
  __shared__ float sAll[256], sTrH[256], sTrG[256];
  const int tid = threadIdx.x;
  const float invN = 1.0f / (float)NROWS;

  float aAll = 0.0f, aTrH = 0.0f, aTrG = 0.0f;
  for (int idx = tid; idx < D * D; idx += 256) aAll += h[idx];
  if (tid < D) { aTrH = h[tid * D + tid]; aTrG = gs[tid * D + tid]; }
  sAll[tid] = aAll; sTrH[tid] = aTrH; sTrG[tid] = aTrG;
  __syncthreads();
  for (int s = 128; s > 0; s >>= 1) {
    if (tid < s) {
      sAll[tid] += sAll[tid + s];
      sTrH[tid] += sTrH[tid + s];
      sTrG[tid] += sTrG[tid + s];
    }
    __syncthreads();
  }
  const float sumAllH = sAll[0], trH = sTrH[0], trG = sTrG[0];

  // grad = (1-k) * offdiag(Gs/N) + diag(k * (diag(Gs)/N - 1))
  for (int idx = tid; idx < D * D; idx += 256) {
    const int i = idx >> 7, j = idx & (D - 1);
    const float g = gs[idx] * invN;
    out[idx] = (i == j) ? KAPPA * (g - 1.0f) : (1.0f - KAPPA) * g;
  }
  if (tid == 0) {
    out[D * D]     = (sumAllH - trH) * invN / (float)(D * D);      // corr_loss
    out[D * D + 1] = (trH - 2.0f * trG) * invN / (float)D + 1.0f;  // whit_loss
  }
}

extern "C" void kernel_launch(void* const* d_in, const int* in_sizes, int n_in,
                              void* d_out, int out_size, void* d_ws, size_t ws_size,
                              hipStream_t stream) {
  const float* x  = (const float*)d_in[0];
  float* gs  = (float*)d_ws;          // 128*128 floats
  float* h   = gs + D * D;            // 128*128 floats
  float* out = (float*)d_out;         // 16384 grad + corr + whit

  decorr_zero_kernel<<<(2 * D * D + 255) / 256, 256, 0, stream>>>(gs, 2 * D * D);
  decorr_gram_kernel<<<NROWS / CHUNK, 256, 0, stream>>>(x, gs, h);
  decorr_finalize_kernel<<<1, 256, 0, stream>>>(gs, h, out);
}